// Illumination_76828374991387
// MI455X (gfx1250) — compile-verified
//
#include <hip/hip_runtime.h>
#include <math.h>

// CDNA5 wave32 WMMA operand types
typedef __attribute__((ext_vector_type(2))) float v2f;   // 16x4 / 4x16 f32 A/B
typedef __attribute__((ext_vector_type(8))) float v8f;   // 16x16 f32 C/D

#define EPS_T   1e-4f
#define BIG     1e10f
#define NRM_EPS 1e-12f

typedef unsigned long long u64;
typedef unsigned int u32;

// Hardware TRANS ops (v_sqrt_f32 / v_rcp_f32) — co-execute with the WMMA pipe
__device__ __forceinline__ float hw_sqrt(float x) { return __builtin_amdgcn_sqrtf(x); }
__device__ __forceinline__ float hw_rcp(float x)  { return __builtin_amdgcn_rcpf(x); }

__device__ __forceinline__ v8f wmma4(v2f a, v2f b, v8f c) {
  // D = A(16x4) x B(4x16) + C ; full fp32 matrix pipe
  return __builtin_amdgcn_wmma_f32_16x16x4_f32(
      /*neg_a=*/false, a, /*neg_b=*/false, b,
      /*c_mod=*/(short)0, c, /*reuse_a=*/false, /*reuse_b=*/false);
}

// Sortable key: (t bits << 32) | sphere_idx. All candidate t are positive, so
// IEEE bits are order-preserving; u64 min == (min t, then min sphere index).
__device__ __forceinline__ u64 pack_key(float t, int idx) {
  return ((u64)__float_as_uint(t) << 32) | (u32)idx;
}
#define BIG_KEY ((((u64)0x501502F9ull) << 32) | 0xFFFFFFFFull)  // bits of 1e10f

__global__ __launch_bounds__(256)
void Illumination_76828374991387_kernel(
    const float* __restrict__ ray_o, const float* __restrict__ ray_d,
    const float* __restrict__ centers, const float* __restrict__ radii,
    const float* __restrict__ light_pos, const int* __restrict__ light_idx,
    float* __restrict__ out, int N)
{
  __shared__ float4 sph[128];     // (cx, cy, cz, |c|^2 - r^2)
  __shared__ u64    ldsKey[256];  // per-ray packed (tmin, argmin) via ds_min_u64

  const int tid = threadIdx.x;

  // ---- stage spheres + init per-ray keys ----
  ldsKey[tid] = BIG_KEY;
  if (tid < 128) {
    float cx = centers[tid * 3 + 0];
    float cy = centers[tid * 3 + 1];
    float cz = centers[tid * 3 + 2];
    float r  = radii[tid];
    float q  = fmaf(cx, cx, fmaf(cy, cy, fmaf(cz, cz, -r * r)));
    sph[tid] = make_float4(cx, cy, cz, q);
  }
  __syncthreads();

  const int  lane = tid & 31;
  const int  wave = tid >> 5;
  const int  mrow = lane & 15;
  const bool hi   = lane >= 16;
  const int  base = blockIdx.x * 256 + wave * 32;   // 32 rays per wave

  // ---- build A matrices for two 16-ray batches ----
  // A layout (16x4 f32): lanes 0-15 -> K=0(v0),K=1(v1); lanes 16-31 -> K=2,K=3
  // b-row: [dx, dy, dz, o.d] ; c-row: [ox, oy, oz, |o|^2]
  auto loadA = [&](int ray, v2f& Ab, v2f& Ac) {
    int rr = ray < N ? ray : N - 1;   // clamp (keeps EXEC all-1s for WMMA)
    float ox = ray_o[rr * 3 + 0], oy = ray_o[rr * 3 + 1], oz = ray_o[rr * 3 + 2];
    float dx = ray_d[rr * 3 + 0], dy = ray_d[rr * 3 + 1], dz = ray_d[rr * 3 + 2];
    float dn  = hw_sqrt(fmaf(dx, dx, fmaf(dy, dy, dz * dz)));
    float inv = hw_rcp(fmaxf(dn, NRM_EPS));
    dx *= inv; dy *= inv; dz *= inv;
    float od = fmaf(ox, dx, fmaf(oy, dy, oz * dz));
    float oo = fmaf(ox, ox, fmaf(oy, oy, oz * oz));
    Ab[0] = hi ? dz : dx;  Ab[1] = hi ? od : dy;
    Ac[0] = hi ? oz : ox;  Ac[1] = hi ? oo : oy;
  };
  v2f AbA, AcA, AbB, AcB;
  loadA(base + mrow,      AbA, AcA);
  loadA(base + 16 + mrow, AbB, AcB);

  u64 keyA[8], keyB[8];
#pragma unroll
  for (int j = 0; j < 8; ++j) { keyA[j] = BIG_KEY; keyB[j] = BIG_KEY; }

  // ---- sweep 8 sphere tiles of 16; 4 WMMAs per tile per wave ----
#pragma unroll
  for (int st = 0; st < 8; ++st) {
    const int scol = st * 16 + mrow;          // this lane's sphere column
    float4 s = sph[scol];
    // B layout (4x16 f32): lanes 0-15 -> K=0(v0),K=1(v1); lanes 16-31 -> K=2,K=3
    v2f Bb, Bc;
    Bb[0] = hi ? -s.z        : -s.x;        Bb[1] = hi ? 1.0f : -s.y;
    Bc[0] = hi ? -2.0f * s.z : -2.0f * s.x; Bc[1] = hi ? 1.0f : -2.0f * s.y;
    v8f zero = {0, 0, 0, 0, 0, 0, 0, 0};
    v8f cq   = {s.w, s.w, s.w, s.w, s.w, s.w, s.w, s.w};  // |c|^2 - r^2 per column

    v8f bA = wmma4(AbA, Bb, zero);   // b = o.d - d.c
    v8f cA = wmma4(AcA, Bc, cq);     // c = |o|^2 - 2 o.c + |c|^2 - r^2
    v8f bB = wmma4(AbB, Bb, zero);
    v8f cB = wmma4(AcB, Bc, cq);

#pragma unroll
    for (int j = 0; j < 8; ++j) {
      // interleaved A/B so independent work fills the TRANS hazard slots
      float ba = bA[j], ca = cA[j];
      float bbv = bB[j], cb = cB[j];
      float discA = fmaf(ba, ba, -ca);
      float discB = fmaf(bbv, bbv, -cb);
      float sqA = hw_sqrt(discA);      // disc<0 -> NaN -> miss via compare
      float sqB = hw_sqrt(discB);
      float tA = -ba - sqA;
      float tB = -bbv - sqB;
      float tselA = (tA > EPS_T) ? tA : BIG;   // NaN compares false
      float tselB = (tB > EPS_T) ? tB : BIG;
      u64 kA = pack_key(tselA, scol);
      u64 kB = pack_key(tselB, scol);
      if (kA < keyA[j]) keyA[j] = kA;          // v_min_u64
      if (kB < keyB[j]) keyB[j] = kB;
    }
  }

  // ---- argmin across the 16 sphere columns: non-returning DS_MIN_U64 ----
  // D-tile rows: lanes 0-15 hold M=j, lanes 16-31 hold M=8+j.
  const int rowOff = wave * 32 + (hi ? 8 : 0);
#pragma unroll
  for (int j = 0; j < 8; ++j) {
    atomicMin(&ldsKey[rowOff + j],      keyA[j]);   // batch A: rays base+0..15
    atomicMin(&ldsKey[rowOff + 16 + j], keyB[j]);   // batch B: rays base+16..31
  }
  __syncthreads();

  // ---- shading: one lane per ray ----
  const int r  = blockIdx.x * 256 + tid;
  const int rc = r < N ? r : N - 1;
  u64 kk = ldsKey[tid];
  float tmin = __uint_as_float((u32)(kk >> 32));
  int   si   = (int)((u32)kk & 127u);
  bool  active = tmin < BIG;

  float ox = ray_o[rc * 3 + 0], oy = ray_o[rc * 3 + 1], oz = ray_o[rc * 3 + 2];
  float dx = ray_d[rc * 3 + 0], dy = ray_d[rc * 3 + 1], dz = ray_d[rc * 3 + 2];
  float dn  = hw_sqrt(fmaf(dx, dx, fmaf(dy, dy, dz * dz)));
  float inv = hw_rcp(fmaxf(dn, NRM_EPS));
  dx *= inv; dy *= inv; dz *= inv;

  float px = fmaf(tmin, dx, ox);
  float py = fmaf(tmin, dy, oy);
  float pz = fmaf(tmin, dz, oz);

  float4 sc = sph[si];
  float nx = px - sc.x, ny = py - sc.y, nz = pz - sc.z;
  float ninv = hw_rcp(fmaxf(hw_sqrt(fmaf(nx, nx, fmaf(ny, ny, nz * nz))), NRM_EPS));
  nx *= ninv; ny *= ninv; nz *= ninv;

  int li = light_idx[rc];
  float lx = light_pos[li * 3 + 0] - px;
  float ly = light_pos[li * 3 + 1] - py;
  float lz = light_pos[li * 3 + 2] - pz;

  // Duff et al. branchless frame; express l in local frame
  float sgn = nz >= 0.0f ? 1.0f : -1.0f;
  float a   = -hw_rcp(sgn + nz);
  float bb  = nx * ny * a;
  float sx = 1.0f + sgn * nx * nx * a, sy = sgn * bb,            szv = -sgn * nx;
  float tx = bb,                        ty = sgn + ny * ny * a,  tz  = -ny;
  float l0 = fmaf(lx, sx, fmaf(ly, sy, lz * szv));
  float l1 = fmaf(lx, tx, fmaf(ly, ty, lz * tz));
  float l2 = fmaf(lx, nx, fmaf(ly, ny, lz * nz));
  float linv = hw_rcp(fmaxf(hw_sqrt(fmaf(l0, l0, fmaf(l1, l1, l2 * l2))), NRM_EPS));
  l0 *= linv; l1 *= linv; l2 *= linv;

  float r0 = active ? (l0 + 1.0f) * 0.5f : 0.0f;
  float r1 = active ? (l1 + 1.0f) * 0.5f : 0.0f;
  float r2 = active ? (l2 + 1.0f) * 0.5f : 0.0f;

  if (r < N) {
    out[r * 3 + 0] = (1.0f + r0) * 0.5f;
    out[r * 3 + 1] = (1.0f + r1) * 0.5f;
    out[r * 3 + 2] = (1.0f + r2) * 0.5f;
    out[(size_t)N * 3 + r] = active ? 1.0f : 0.0f;
  }
}

extern "C" void kernel_launch(void* const* d_in, const int* in_sizes, int n_in,
                              void* d_out, int out_size, void* d_ws, size_t ws_size,
                              hipStream_t stream) {
  const float* ray_o     = (const float*)d_in[0];
  const float* ray_d     = (const float*)d_in[1];
  const float* centers   = (const float*)d_in[2];
  const float* radii     = (const float*)d_in[3];
  const float* light_pos = (const float*)d_in[4];
  const int*   light_idx = (const int*)d_in[5];
  float* out = (float*)d_out;
  int N = in_sizes[0] / 3;
  int blocks = (N + 255) / 256;
  Illumination_76828374991387_kernel<<<blocks, 256, 0, stream>>>(
      ray_o, ray_d, centers, radii, light_pos, light_idx, out, N);
}